// NeuralBellmanFordNetwork_20435454394377
// MI455X (gfx1250) — compile-verified
//
#include <hip/hip_runtime.h>

// NBFNet forward, MI455X (gfx1250, wave32).
// Phase structure per layer:
//   1) agg = boundary (zero + 128-thread scatter)
//   2) edge kernel: wave-per-edge DistMult message + global_atomic_add_f32 scatter
//   3) node update GEMM [N*B,64]@[64,32] via V_WMMA_F32_16X16X4_F32, bias+ReLU fused,
//      written in place over agg; buffers ping-pong.
// Final scorer evaluated only at the 132 gathered (t_index, b) pairs.

#define N_NODE  50000
#define N_REL   20
#define DIM     32
#define N_LAYER 4
#define NEDGE   200000
#define BB      4
#define NEG     33

typedef float v2f __attribute__((ext_vector_type(2)));
typedef float v8f __attribute__((ext_vector_type(8)));

__global__ void zero_kernel(float4* __restrict__ p, long n4) {
    long i = (long)blockIdx.x * blockDim.x + threadIdx.x;
    long stride = (long)gridDim.x * blockDim.x;
    float4 z; z.x = z.y = z.z = z.w = 0.f;
    for (; i < n4; i += stride) p[i] = z;
}

// dst[h_index[b], b, :] = query_emb[r_index[b], :]   (launch: 1 block x 128)
__global__ void boundary_kernel(float* __restrict__ dst,
                                const int* __restrict__ h_index,
                                const int* __restrict__ r_index,
                                const float* __restrict__ query_emb) {
    int b = threadIdx.x >> 5;
    int d = threadIdx.x & 31;
    if (b < BB)
        dst[(long)h_index[b] * (BB * DIM) + b * DIM + d] =
            query_emb[r_index[b] * DIM + d];
}

// One wave per (possibly inverted) edge; lane = feature dim (DIM==warpSize==32).
__global__ void edge_kernel(const float* __restrict__ x, float* __restrict__ agg,
                            const float* __restrict__ rel,          // [2*N_REL, DIM] (layer slice)
                            const int* __restrict__ edge_index,     // [E,2]
                            const int* __restrict__ edge_type,      // [E]
                            const float* __restrict__ edge_weight)  // [E]
{
    int wid  = (blockIdx.x * blockDim.x + threadIdx.x) >> 5;
    int lane = threadIdx.x & 31;
    if (wid >= 2 * NEDGE) return;
    int s, t, rt; float w;
    if (wid < NEDGE) {
        s = edge_index[2 * wid];     t = edge_index[2 * wid + 1];
        rt = edge_type[wid];         w = edge_weight[wid];
    } else {
        int e2 = wid - NEDGE;
        s = edge_index[2 * e2 + 1];  t = edge_index[2 * e2];
        rt = edge_type[e2] + N_REL;  w = edge_weight[e2];
    }
    float r = rel[rt * DIM + lane] * w;
    const float* xs = x   + (long)s * (BB * DIM) + lane;
    float*       at = agg + (long)t * (BB * DIM) + lane;
#pragma unroll
    for (int b = 0; b < BB; ++b)
        atomicAdd(at + b * DIM, xs[b * DIM] * r);     // global_atomic_add_f32, coalesced
}

// y[r,:] = relu(cat([agg[r,:], x[r,:]]) @ W[64,32] + bias), y aliases agg (in place).
// One wave handles 16 rows x 32 cols via 2 accumulator tiles of V_WMMA_F32_16X16X4_F32.
__global__ void layer_gemm_kernel(float* __restrict__ agg,       // feat[:, 0:32] and output
                                  const float* __restrict__ x,   // feat[:, 32:64]
                                  const float* __restrict__ W,   // [64,32] row-major
                                  const float* __restrict__ bias,// [32]
                                  int rows)
{
    int wid  = (blockIdx.x * blockDim.x + threadIdx.x) >> 5;
    int lane = threadIdx.x & 31;
    int row0 = wid * 16;
    if (row0 >= rows) return;               // wave-uniform: EXEC all-1s inside WMMA

    int half = lane >> 4;                   // K sub-pair select (ISA 16x4 f32 A layout)
    int l16  = lane & 15;
    const float* aggRow = agg + (long)(row0 + l16) * DIM;
    const float* xRow   = x   + (long)(row0 + l16) * DIM;

    v8f acc0 = {};  // cols 0..15
    v8f acc1 = {};  // cols 16..31

#pragma unroll
    for (int k = 0; k < 2 * DIM; k += 4) {
        int k0 = k + 2 * half;              // this lane's K pair: {k0, k0+1}, same source half
        const float* ap = (k0 < DIM) ? (aggRow + k0) : (xRow + (k0 - DIM));
        v2f a; a.x = ap[0]; a.y = ap[1];    // 8B-aligned float2

        const float* w0 = W + (long)k0 * DIM;   // B[k'][n] = W[k0+k'][ntile*16 + l16]
        v2f b0; b0.x = w0[l16];      b0.y = w0[DIM + l16];
        v2f b1; b1.x = w0[16 + l16]; b1.y = w0[DIM + 16 + l16];

        acc0 = __builtin_amdgcn_wmma_f32_16x16x4_f32(false, a, false, b0,
                                                     (short)0, acc0, false, false);
        acc1 = __builtin_amdgcn_wmma_f32_16x16x4_f32(false, a, false, b1,
                                                     (short)0, acc1, false, false);
    }

    // C/D layout: vgpr j -> row row0+j (lanes 0..15) / row0+j+8 (lanes 16..31), col = l16 (+16)
    float bl = bias[l16], bh = bias[16 + l16];
#pragma unroll
    for (int j = 0; j < 8; ++j) {
        long r = row0 + j + 8 * half;
        float v0 = acc0[j] + bl;
        float v1 = acc1[j] + bh;
        agg[r * DIM + l16]      = v0 > 0.f ? v0 : 0.f;
        agg[r * DIM + 16 + l16] = v1 > 0.f ? v1 : 0.f;
    }
}

// Final MLP only at the gathered (b, t_index[b,j]) pairs: one wave per output.
__global__ void score_kernel(const float* __restrict__ xfin,
                             const int* __restrict__ t_index,   // [B*NEG]
                             const int* __restrict__ r_index,
                             const float* __restrict__ query_emb,
                             const float* __restrict__ W1,      // [64,64]
                             const float* __restrict__ b1,      // [64]
                             const float* __restrict__ W2,      // [64]
                             const float* __restrict__ b2,      // [1]
                             float* __restrict__ out)           // [B*NEG]
{
    int wid  = (blockIdx.x * blockDim.x + threadIdx.x) >> 5;
    int lane = threadIdx.x & 31;
    if (wid >= BB * NEG) return;
    int b = wid / NEG;
    int t = t_index[wid];
    const float* xr = xfin + (long)t * (BB * DIM) + b * DIM;
    const float* q  = query_emb + r_index[b] * DIM;

    float partial = 0.f;
#pragma unroll
    for (int uu = 0; uu < 2; ++uu) {
        int u = lane + uu * 32;
        float h = b1[u];
        for (int k = 0; k < DIM; ++k) h += xr[k] * W1[k * (2 * DIM) + u];
        for (int k = 0; k < DIM; ++k) h += q[k]  * W1[(DIM + k) * (2 * DIM) + u];
        h = h > 0.f ? h : 0.f;
        partial += h * W2[u];
    }
    for (int off = 16; off > 0; off >>= 1)
        partial += __shfl_xor(partial, off, 32);
    if (lane == 0) out[wid] = partial + b2[0];
}

extern "C" void kernel_launch(void* const* d_in, const int* in_sizes, int n_in,
                              void* d_out, int out_size, void* d_ws, size_t ws_size,
                              hipStream_t stream) {
    const int*   edge_index  = (const int*)  d_in[0];
    const int*   edge_type   = (const int*)  d_in[1];
    const float* edge_weight = (const float*)d_in[2];
    const int*   h_index     = (const int*)  d_in[3];
    const int*   t_index     = (const int*)  d_in[4];
    const int*   r_index     = (const int*)  d_in[5];
    const float* query_emb   = (const float*)d_in[6];
    const float* rel_emb     = (const float*)d_in[7];
    const float* layer_W     = (const float*)d_in[8];
    const float* layer_b     = (const float*)d_in[9];
    const float* mlp_W1      = (const float*)d_in[10];
    const float* mlp_b1      = (const float*)d_in[11];
    const float* mlp_W2      = (const float*)d_in[12];
    const float* mlp_b2      = (const float*)d_in[13];
    float* out = (float*)d_out;

    const long nfeat = (long)N_NODE * BB * DIM;   // 6.4M floats (25.6 MB)
    float* x   = (float*)d_ws;
    float* agg = x + nfeat;

    const long n4 = nfeat / 4;
    const int  rows = N_NODE * BB;                         // 200000 (multiple of 16)
    const int  gemm_blocks = ((rows / 16) * 32 + 255) / 256;
    const int  edge_blocks = (2 * NEDGE) / 8;              // 8 waves/block, exact

    zero_kernel<<<1024, 256, 0, stream>>>((float4*)x, n4);
    boundary_kernel<<<1, 128, 0, stream>>>(x, h_index, r_index, query_emb);

    for (int i = 0; i < N_LAYER; ++i) {
        zero_kernel<<<1024, 256, 0, stream>>>((float4*)agg, n4);
        boundary_kernel<<<1, 128, 0, stream>>>(agg, h_index, r_index, query_emb);
        edge_kernel<<<edge_blocks, 256, 0, stream>>>(
            x, agg, rel_emb + (long)i * 2 * N_REL * DIM,
            edge_index, edge_type, edge_weight);
        layer_gemm_kernel<<<gemm_blocks, 256, 0, stream>>>(
            agg, x, layer_W + (long)i * 2 * DIM * DIM, layer_b + i * DIM, rows);
        float* tmp = x; x = agg; agg = tmp;
    }

    score_kernel<<<(BB * NEG * 32 + 127) / 128, 128, 0, stream>>>(
        x, t_index, r_index, query_emb, mlp_W1, mlp_b1, mlp_W2, mlp_b2, out);
}